// GridEncodingBase_86036784874051
// MI455X (gfx1250) — compile-verified
//
#include <hip/hip_runtime.h>
#include <hip/hip_bf16.h>

typedef __attribute__((ext_vector_type(2))) float v2f;
typedef __attribute__((ext_vector_type(8))) float v8f;

// ---------------------------------------------------------------------------
// Stage 1: H_i = reshape(g_i, (R_i*4, Cg)) @ lin_w[:, off:off+Cg]^T
// fp32 WMMA 16x16x4. One wave per 16x16 output tile; 8 waves/block cover the
// 128 output channels; blockIdx.x covers row tiles.
// ---------------------------------------------------------------------------
__global__ __launch_bounds__(256) void ge_gemm_f32_wmma(
    const float* __restrict__ G,      // (M, K) row-major (view of grid)
    const float* __restrict__ Wlin,   // (128, 448) row-major
    float* __restrict__ Hout,         // (M, 128) row-major
    int K, int off) {
  const int lane = threadIdx.x & 31;
  const int wave = threadIdx.x >> 5;        // column tile 0..7
  const int rt   = blockIdx.x;              // row tile
  const int l15  = lane & 15;
  const int kg   = 2 * (lane >> 4);         // K-pair base within a 4-step

  // A: lane holds {G[m][kb+kg], G[m][kb+kg+1]},  m = rt*16 + l15
  const float* arow = G + (size_t)(rt * 16 + l15) * K + kg;
  // B: lane holds {W[n][off+kb+kg], W[n][off+kb+kg+1]}, n = wave*16 + l15
  const float* brow = Wlin + (size_t)(wave * 16 + l15) * 448 + off + kg;

  v8f acc = {};
  for (int kb = 0; kb < K; kb += 4) {
    v2f a = *(const v2f*)(arow + kb);
    v2f b = *(const v2f*)(brow + kb);
    acc = __builtin_amdgcn_wmma_f32_16x16x4_f32(
        /*neg_a=*/false, a, /*neg_b=*/false, b,
        /*c_mod=*/(short)0, acc, /*reuse_a=*/false, /*reuse_b=*/false);
  }

  // D: VGPR r -> row (8*(lane>>4) + r), col (lane&15)
  float* drow = Hout + (size_t)(rt * 16 + 8 * (lane >> 4)) * 128 + wave * 16 + l15;
#pragma unroll
  for (int r = 0; r < 8; ++r) drow[(size_t)r * 128] = acc[r];
}

// ---------------------------------------------------------------------------
// Stage 2: per-(n,t,h) workgroup. Blend t & h at LDS-stage time, then sweep w.
// ---------------------------------------------------------------------------
__device__ __forceinline__ void ge_prep_axis(float c, int S, int& i0, int& i1, float& f) {
  float x  = (c + 1.0f) * 0.5f * (float)(S - 1);
  float x0 = floorf(x);
  x0 = fminf(fmaxf(x0, 0.0f), (float)(S - 1));
  i0 = (int)x0;
  i1 = i0 + 1;
  if (i1 > S - 1) i1 = S - 1;
  f = x - x0;
}

__global__ __launch_bounds__(512) void ge_interp(
    const float* __restrict__ coor_t,  // (2,12)
    const float* __restrict__ coor_h,  // (2,36)
    const float* __restrict__ coor_w,  // (2,64)
    const float* __restrict__ H0,      // (17280*4, 128) == rows (Tg*9*16) x 512
    const float* __restrict__ H1,
    const float* __restrict__ H2,
    const float* __restrict__ lin_b,   // (128,)
    float* __restrict__ out) {         // (2,12,36,64,512)
  __shared__ float lds[3 * 16 * 512];  // 96 KB: 3 grids x 16 wi-cells x 512 ch

  const int tid = threadIdx.x;
  const int bid = blockIdx.x;          // (n*12 + t)*36 + h
  const int n   = bid / (12 * 36);
  const int rem = bid % (12 * 36);
  const int t   = rem / 36;
  const int h   = rem % 36;

  const float ctv = coor_t[n * 12 + t];
  const float chv = coor_h[n * 36 + h];

  int hi0, hi1; float fh;
  ge_prep_axis(chv, 9, hi0, hi1, fh);  // Hg==9 for all grids

  const float* Hs[3]  = {H0, H1, H2};
  const int    Tgs[3] = {120, 60, 30};

#pragma unroll
  for (int i = 0; i < 3; ++i) {
    int ti0, ti1; float ft;
    ge_prep_axis(ctv, Tgs[i], ti0, ti1, ft);
    const float* Hp = Hs[i];
    // row(tt,hh,wi) = ((tt*9 + hh)*16 + wi); element stride 512 (=4*128 ch)
    const size_t b00 = (size_t)(ti0 * 9 + hi0) * 16 * 512;
    const size_t b01 = (size_t)(ti0 * 9 + hi1) * 16 * 512;
    const size_t b10 = (size_t)(ti1 * 9 + hi0) * 16 * 512;
    const size_t b11 = (size_t)(ti1 * 9 + hi1) * 16 * 512;
    for (int wi = 0; wi < 16; ++wi) {
      const size_t o = (size_t)wi * 512 + tid;
      float v00 = Hp[b00 + o];
      float v01 = Hp[b01 + o];
      float v10 = Hp[b10 + o];
      float v11 = Hp[b11 + o];
      float lo = v00 + (v01 - v00) * fh;
      float hi = v10 + (v11 - v10) * fh;
      lds[(i * 16 + wi) * 512 + tid] = lo + (hi - lo) * ft;
    }
  }
  __syncthreads();

  const float bv = lin_b[tid & 127];
  float* orow = out + (size_t)bid * 64 * 512 + tid;
  for (int w = 0; w < 64; ++w) {
    const float cwv = coor_w[n * 64 + w];
    int wi0, wi1; float fw;
    ge_prep_axis(cwv, 16, wi0, wi1, fw);   // Wg==16 for all grids
    float acc = bv;
#pragma unroll
    for (int i = 0; i < 3; ++i) {
      float a = lds[(i * 16 + wi0) * 512 + tid];
      float b = lds[(i * 16 + wi1) * 512 + tid];
      acc += a + (b - a) * fw;
    }
    orow[(size_t)w * 512] = acc;
  }
}

// ---------------------------------------------------------------------------
extern "C" void kernel_launch(void* const* d_in, const int* in_sizes, int n_in,
                              void* d_out, int out_size, void* d_ws, size_t ws_size,
                              hipStream_t stream) {
  const float* coor_t = (const float*)d_in[0];
  const float* coor_h = (const float*)d_in[1];
  const float* coor_w = (const float*)d_in[2];
  const float* g0     = (const float*)d_in[3];  // (17280, 256)
  const float* g1     = (const float*)d_in[4];  // (8640, 512)
  const float* g2     = (const float*)d_in[5];  // (4320, 1024)
  const float* lin_w  = (const float*)d_in[6];  // (128, 448)
  const float* lin_b  = (const float*)d_in[7];  // (128,)
  float* out = (float*)d_out;

  // Workspace layout: H0 | H1 | H2  (total 61,931,520 bytes)
  float* H0 = (float*)d_ws;                       // 69120 x 128
  float* H1 = H0 + (size_t)69120 * 128;           // 34560 x 128
  float* H2 = H1 + (size_t)34560 * 128;           // 17280 x 128

  // Stage 1: three small fp32-WMMA GEMMs (channel slices of lin_w)
  // g0 viewed as (17280*4, 64), g1 as (8640*4, 128), g2 as (4320*4, 256)
  ge_gemm_f32_wmma<<<69120 / 16, 256, 0, stream>>>(g0, lin_w, H0, 64, 0);
  ge_gemm_f32_wmma<<<34560 / 16, 256, 0, stream>>>(g1, lin_w, H1, 128, 64);
  ge_gemm_f32_wmma<<<17280 / 16, 256, 0, stream>>>(g2, lin_w, H2, 256, 192);

  // Stage 2: trilinear blend of the pre-projected grids + bias
  ge_interp<<<2 * 12 * 36, 512, 0, stream>>>(coor_t, coor_h, coor_w,
                                             H0, H1, H2, lin_b, out);
}